// Residual_74629351735837
// MI455X (gfx1250) — compile-verified
//
#include <hip/hip_runtime.h>
#include <stdint.h>

// Bundle-adjustment residual: HBM-bandwidth-bound elementwise kernel.
// CDNA5 feature used: Tensor Data Mover DMA of the 56KB pose table into LDS
// (one tensor_load_to_lds per block), synchronized with s_wait_tensorcnt.

#define NUM_CAM 2000
#define POSE_F  (NUM_CAM * 7)     // 14000 floats
#define NBLOCKS 1024
#define NTHREADS 256

typedef unsigned int u32x4 __attribute__((ext_vector_type(4)));
typedef int          i32x8 __attribute__((ext_vector_type(8)));
typedef int          i32x4 __attribute__((ext_vector_type(4)));

__global__ __launch_bounds__(NTHREADS) void ba_residual_kernel(
    const float* __restrict__ poses,      // (C,7)
    const float* __restrict__ pts,        // (P,3)
    const float* __restrict__ obs,        // (P,2)
    const float* __restrict__ Km,         // (3,3)
    const int*   __restrict__ cidx,       // (P,)
    const int*   __restrict__ pidx,       // (P,)
    float*       __restrict__ out,        // (P,2)
    int P)
{
    __shared__ float s_poses[POSE_F];     // 56000 bytes

    // ---- Stage full pose table into LDS with one TDM DMA (wave 0 only) ----
    if (threadIdx.x < 32u) {              // wave-uniform branch
        uint64_t ga  = (uint64_t)(uintptr_t)poses;
        uint32_t lds = (uint32_t)(uintptr_t)(&s_poses[0]);   // low 32 bits = LDS byte offset

        u32x4 g0;
        g0[0] = 1u;                                   // count=1, user mode, no gather
        g0[1] = lds;                                  // lds_addr
        g0[2] = (uint32_t)(ga & 0xFFFFFFFFu);         // global_addr[31:0]
        g0[3] = (uint32_t)((ga >> 32) & 0x01FFFFFFu)  // global_addr[56:32]
              | (2u << 30);                           // D# type = 2 ("image")

        i32x8 g1;
        g1[0] = (int)(2u << 16);                      // workgroup_mask=0, data_size=2 (4B)
        g1[1] = (int)((uint32_t)(POSE_F & 0xFFFF) << 16);   // tensor_dim0[15:0]
        g1[2] = (int)(((POSE_F >> 16) & 0xFFFF)             // tensor_dim0[31:16]
              | (1u << 16));                                // tensor_dim1 = 1
        g1[3] = (int)((uint32_t)(POSE_F & 0xFFFF) << 16);   // tile_dim0 = 14000
        g1[4] = 1;                                    // tile_dim1 = 1, tile_dim2 = 0
        g1[5] = POSE_F;                               // tensor_dim0_stride[31:0]
        g1[6] = 0;                                    // stride hi, dim1_stride lo
        g1[7] = 0;

        i32x4 g2 = (i32x4){0, 0, 0, 0};               // 2D tensor: groups 2/3 unused
        i32x4 g3 = (i32x4){0, 0, 0, 0};
        i32x8 g4 = (i32x8){0, 0, 0, 0, 0, 0, 0, 0};   // extra group (clang-23 6-arg form)

        __builtin_amdgcn_tensor_load_to_lds(g0, g1, g2, g3, g4, 0);
        __builtin_amdgcn_s_wait_tensorcnt(0);
    }
    __syncthreads();

    // Intrinsics (uniform scalar loads)
    const float fx  = Km[0];
    const float cxk = Km[2];
    const float fy  = Km[4];
    const float cyk = Km[5];

    const int stride = (int)(gridDim.x * blockDim.x);
    for (int i = (int)(blockIdx.x * blockDim.x + threadIdx.x); i < P; i += stride) {
        const int ci = cidx[i];
        const int pi = pidx[i];

        const float* pp = &s_poses[ci * 7];           // LDS gather
        const float tx = pp[0], ty = pp[1], tz = pp[2];
        const float qx = pp[3], qy = pp[4], qz = pp[5], qw = pp[6];

        const float* pv = &pts[(size_t)pi * 3];       // coalesced b96
        const float px = pv[0], py = pv[1], pz = pv[2];

        // uv = cross(qv, p) + qw * p
        const float ux = qy * pz - qz * py + qw * px;
        const float uy = qz * px - qx * pz + qw * py;
        const float uz = qx * py - qy * px + qw * pz;

        // pc = p + 2*cross(qv, uv) + t
        const float X = px + 2.0f * (qy * uz - qz * uy) + tx;
        const float Y = py + 2.0f * (qz * ux - qx * uz) + ty;
        const float Z = pz + 2.0f * (qx * uy - qy * ux) + tz;

        // projection: proj = pc @ K.T ; pix = proj[:2]/proj[2]
        const float pr0 = fx * X + cxk * Z;
        const float pr1 = fy * Y + cyk * Z;

        const float2 ob = ((const float2*)obs)[i];    // coalesced b64
        float2 r;
        r.x = pr0 / Z - ob.x;
        r.y = pr1 / Z - ob.y;
        ((float2*)out)[i] = r;                        // coalesced b64
    }
}

extern "C" void kernel_launch(void* const* d_in, const int* in_sizes, int n_in,
                              void* d_out, int out_size, void* d_ws, size_t ws_size,
                              hipStream_t stream) {
    const float* poses = (const float*)d_in[0];
    const float* pts   = (const float*)d_in[1];
    const float* obs   = (const float*)d_in[2];
    const float* K     = (const float*)d_in[3];
    const int*   cidx  = (const int*)d_in[4];
    const int*   pidx  = (const int*)d_in[5];
    float* out = (float*)d_out;
    const int P = in_sizes[4];   // element count of cidx == P

    ba_residual_kernel<<<dim3(NBLOCKS), dim3(NTHREADS), 0, stream>>>(
        poses, pts, obs, K, cidx, pidx, out, P);
}